// BipartiteEncoder_63857573757446
// MI455X (gfx1250) — compile-verified
//
#include <hip/hip_runtime.h>
#include <hip/hip_bf16.h>

typedef __attribute__((ext_vector_type(2))) float v2f;
typedef __attribute__((ext_vector_type(8))) float v8f;

#define DIM 128
#define WT_STRIDE 132   // transposed-W LDS row stride (floats); 132 % 64 == 4 -> conflict-free b64 reads
#define NEG_SLOPE 0.01f

// ---------------- degree / normalization precompute ----------------

__global__ __launch_bounds__(256) void init_deg_kernel(float* __restrict__ deg, int n) {
    int i = blockIdx.x * blockDim.x + threadIdx.x;
    if (i < n) deg[i] = 1.0f;   // self-loop weight
}

__global__ __launch_bounds__(256) void deg_scatter_kernel(const long long* __restrict__ ei,
                                                          const float* __restrict__ ew,
                                                          float* __restrict__ deg, int E) {
    int e = blockIdx.x * blockDim.x + threadIdx.x;
    if (e < E) {
        long long c = ei[(size_t)E + e];   // col = edge_index[1]
        atomicAdd(&deg[c], ew[e]);
    }
}

__global__ __launch_bounds__(256) void dinv_kernel(float* __restrict__ deg, int n) {
    int i = blockIdx.x * blockDim.x + threadIdx.x;
    if (i < n) {
        float d = deg[i];
        deg[i] = (d > 0.0f) ? rsqrtf(d) : 0.0f;
    }
}

// ---------------- fused dual GEMM (WMMA f32 16x16x4) + epilogue ----------------
// 512 threads = 16 waves. Waves 0-7: rows strips 0-7, cols 0-63. Waves 8-15: same strips, cols 64-127.
// Per wave: 16 rows x 64 cols for BOTH GEMMs -> 8 v8f accumulators (64 VGPRs).
// h = x @ W_gcn                       -> ws
// out = leaky(x @ W_fnn + b_fnn) + b_gcn + h * dinv^2   (self-loop message folded in)

__global__ __launch_bounds__(512) void gemm_fused_kernel(
    const float* __restrict__ x_src, const float* __restrict__ x_tar,
    const float* __restrict__ Wg, const float* __restrict__ Wf,
    const float* __restrict__ bg, const float* __restrict__ bf,
    const float* __restrict__ dinv,
    float* __restrict__ h, float* __restrict__ out,
    int n_src, int n_total)
{
    // W stored TRANSPOSED in LDS: sW*[c * WT_STRIDE + k] = W[k][c]
    __shared__ float sWg[DIM * WT_STRIDE];
    __shared__ float sWf[DIM * WT_STRIDE];

    const int t = threadIdx.x;
    for (int i = t; i < DIM * DIM; i += 512) {
        int k = i >> 7, c = i & 127;
        sWg[c * WT_STRIDE + k] = Wg[i];
        sWf[c * WT_STRIDE + k] = Wf[i];
    }
    __syncthreads();

    const int lane  = t & 31;
    const int wave  = t >> 5;
    const int strip = wave & 7;        // row strip within the 128-row block
    const int nh    = wave >> 3;       // column half: 0 -> cols 0-63, 1 -> cols 64-127
    const int m0    = blockIdx.x * 128 + strip * 16;

    // A fragment addressing per ISA 16x4 f32 layout:
    // lane L holds row M = L%16; VGPR j holds K = 4*s + j + 2*(L>=16)
    const int mrow = m0 + (lane & 15);
    const int kshift = (lane >> 4) * 2;
    const int mclamp = (mrow < n_total) ? mrow : (n_total - 1);
    const float* arow = ((mclamp < n_src) ? (x_src + (size_t)mclamp * DIM)
                                          : (x_tar + (size_t)(mclamp - n_src) * DIM)) + kshift;
    const int cidx = lane & 15;

    // Loop-invariant LDS base indices: the 16*s byte term folds into the ds immediate offset.
    int cb[4];
#pragma unroll
    for (int j = 0; j < 4; ++j)
        cb[j] = (nh * 64 + j * 16 + cidx) * WT_STRIDE + kshift;

    v8f accg[4], accf[4];
#pragma unroll
    for (int i = 0; i < 4; ++i) { v8f z = {}; accg[i] = z; accf[i] = z; }

#pragma unroll 4
    for (int s = 0; s < 32; ++s) {
        v2f a = *(const v2f*)(arow + 4 * s);              // global_load_b64, imm offset
#pragma unroll
        for (int j = 0; j < 4; ++j) {
            v2f bG = *(const v2f*)&sWg[cb[j] + 4 * s];    // ds_load_b64, imm offset
            v2f bF = *(const v2f*)&sWf[cb[j] + 4 * s];
            accg[j] = __builtin_amdgcn_wmma_f32_16x16x4_f32(false, a, false, bG,
                                                            (short)0, accg[j], false, false);
            accf[j] = __builtin_amdgcn_wmma_f32_16x16x4_f32(false, a, false, bF,
                                                            (short)0, accf[j], false, false);
        }
    }

    // C/D layout: VGPR v, lanes 0-15 -> row m0+v, lanes 16-31 -> row m0+v+8; col = c0 + lane%16
    const int rbase = m0 + ((lane >> 4) * 8);
    float dv2[8];
#pragma unroll
    for (int v = 0; v < 8; ++v) {
        int r = rbase + v;
        float d = (r < n_total) ? dinv[r] : 0.0f;
        dv2[v] = d * d;
    }
    float bgv[4], bfv[4];
#pragma unroll
    for (int j = 0; j < 4; ++j) {
        bgv[j] = bg[nh * 64 + j * 16 + cidx];
        bfv[j] = bf[nh * 64 + j * 16 + cidx];
    }

#pragma unroll
    for (int v = 0; v < 8; ++v) {
        int r = rbase + v;
        if (r < n_total) {
#pragma unroll
            for (int j = 0; j < 4; ++j) {
                int c = nh * 64 + j * 16 + cidx;
                float hv = accg[j][v];
                float fv = accf[j][v] + bfv[j];
                fv = (fv > 0.0f) ? fv : NEG_SLOPE * fv;
                h[(size_t)r * DIM + c] = hv;
                out[(size_t)r * DIM + c] = fv + bgv[j] + hv * dv2[v];
            }
        }
    }
}

// ---------------- edge scatter: one wave per edge ----------------

__global__ __launch_bounds__(256) void edge_scatter_kernel(
    const long long* __restrict__ ei, const float* __restrict__ ew,
    const float* __restrict__ dinv, const float* __restrict__ h,
    float* __restrict__ out, int E)
{
    const int gw = (int)((blockIdx.x * (unsigned)blockDim.x + threadIdx.x) >> 5);
    const int lane = threadIdx.x & 31;
    if (gw >= E) return;   // wave-uniform
    long long r = ei[gw];
    long long c = ei[(size_t)E + gw];
    float coef = dinv[r] * ew[gw] * dinv[c];
    const float4 hv = *(const float4*)(h + (size_t)r * DIM + lane * 4);
    float* op = out + (size_t)c * DIM + lane * 4;
    atomicAdd(op + 0, coef * hv.x);
    atomicAdd(op + 1, coef * hv.y);
    atomicAdd(op + 2, coef * hv.z);
    atomicAdd(op + 3, coef * hv.w);
}

// ---------------- per-row normalization + leaky_relu (one wave per row) ----------------

__global__ __launch_bounds__(256) void norm_kernel(float* __restrict__ out, int n)
{
    const int gw = (int)((blockIdx.x * (unsigned)blockDim.x + threadIdx.x) >> 5);
    const int lane = threadIdx.x & 31;
    if (gw >= n) return;   // wave-uniform
    float* p = out + (size_t)gw * DIM + lane * 4;
    float4 v = *(float4*)p;
    float s = v.x + v.y + v.z + v.w;
#pragma unroll
    for (int off = 16; off > 0; off >>= 1) s += __shfl_xor(s, off, 32);
    const float mean = s * (1.0f / 128.0f);
    float dx = v.x - mean, dy = v.y - mean, dz = v.z - mean, dw = v.w - mean;
    float q = dx * dx + dy * dy + dz * dz + dw * dw;
#pragma unroll
    for (int off = 16; off > 0; off >>= 1) q += __shfl_xor(q, off, 32);
    const float inv = rsqrtf(q * (1.0f / 128.0f) + 1e-6f);
    float4 o;
    o.x = dx * inv; o.y = dy * inv; o.z = dz * inv; o.w = dw * inv;
    o.x = (o.x > 0.0f) ? o.x : NEG_SLOPE * o.x;
    o.y = (o.y > 0.0f) ? o.y : NEG_SLOPE * o.y;
    o.z = (o.z > 0.0f) ? o.z : NEG_SLOPE * o.z;
    o.w = (o.w > 0.0f) ? o.w : NEG_SLOPE * o.w;
    *(float4*)p = o;
}

// ---------------- launcher ----------------

extern "C" void kernel_launch(void* const* d_in, const int* in_sizes, int n_in,
                              void* d_out, int out_size, void* d_ws, size_t ws_size,
                              hipStream_t stream) {
    const float*     x_src = (const float*)d_in[0];
    const float*     x_tar = (const float*)d_in[1];
    const long long* ei    = (const long long*)d_in[2];
    const float*     ew    = (const float*)d_in[3];
    const float*     Wg    = (const float*)d_in[4];
    const float*     bg    = (const float*)d_in[5];
    const float*     Wf    = (const float*)d_in[6];
    const float*     bf    = (const float*)d_in[7];

    const int n_src = in_sizes[0] / DIM;
    const int n_tar = in_sizes[1] / DIM;
    const int n     = n_src + n_tar;
    const int E     = in_sizes[3];

    float* out  = (float*)d_out;
    float* dinv = (float*)d_ws;                       // n floats (deg -> dinv in place)
    float* h    = dinv + (((size_t)n + 3) & ~(size_t)3);  // n*DIM floats

    init_deg_kernel<<<(n + 255) / 256, 256, 0, stream>>>(dinv, n);
    deg_scatter_kernel<<<(E + 255) / 256, 256, 0, stream>>>(ei, ew, dinv, E);
    dinv_kernel<<<(n + 255) / 256, 256, 0, stream>>>(dinv, n);

    gemm_fused_kernel<<<(n + 127) / 128, 512, 0, stream>>>(
        x_src, x_tar, Wg, Wf, bg, bf, dinv, h, out, n_src, n);

    edge_scatter_kernel<<<(E + 7) / 8, 256, 0, stream>>>(ei, ew, dinv, h, out, E);

    norm_kernel<<<(n + 7) / 8, 256, 0, stream>>>(out, n);
}